// ScaledDotProductAttention_42090679501324
// MI455X (gfx1250) — compile-verified
//
#include <hip/hip_runtime.h>

typedef __attribute__((ext_vector_type(2))) float v2f;
typedef __attribute__((ext_vector_type(4))) float v4f;
typedef __attribute__((ext_vector_type(8))) float v8f;
typedef __attribute__((ext_vector_type(4))) int   v4i;

#define B_   2
#define H_   16
#define S_   2048
#define DK_  64
#define QT   16                  // query rows per workgroup
#define NT_TILES (S_ / 16)       // 128 key column tiles
#define LDS_STRIDE (S_ + 4)      // 2052 floats/row (16B-aligned rows, bank-skewed)
#define RED_STRIDE 17            // padded reduction tile stride
#define RED_FLOATS (4 * 16 * RED_STRIDE)   // 4 tiles x 16 rows x 17

// One workgroup = one (b,h) x 16-query tile.
// Phase 1: scores = (Q*scale) K^T via V_WMMA_F32_16X16X4_F32 -> LDS (128 KB)
// Phase 2: mask + row softmax with the row held in VGPRs (16 x float4/lane),
//          single LDS load pass + single LDS store pass, NT b128 attention store
// Phase 3: out = P V via WMMA, split-K over all 8 waves, LDS reduction
__global__ __launch_bounds__(256)
void ScaledDotProductAttention_42090679501324_kernel(
    const float* __restrict__ Q, const float* __restrict__ K,
    const float* __restrict__ V, const int* __restrict__ M,
    float* __restrict__ Out, float* __restrict__ Attn)
{
    extern __shared__ float smem[];          // [QT][LDS_STRIDE] + reduction region

    const int tid  = threadIdx.x;
    const int wave = __builtin_amdgcn_readfirstlane(tid >> 5);  // uniform 0..7
    const int lane = tid & 31;
    const int ln   = lane & 15;              // N / M selector within half-wave
    const int hi   = lane >> 4;              // half-wave selector

    const int qt    = blockIdx.x & (NT_TILES - 1);   // 0..127
    const int bh    = blockIdx.x >> 7;               // 0..31
    const int b     = bh >> 4;                       // H_ == 16
    const int qbase = qt * QT;

    const float* q = Q + ((size_t)bh * S_ + qbase) * DK_;
    const float* k = K + (size_t)bh * S_ * DK_;
    const float* v = V + (size_t)bh * S_ * DK_;
    const int*   m = M + ((size_t)b  * S_ + qbase) * S_;   // mask broadcast over H

    // ---------------- Phase 1: raw scores -> LDS ----------------
    // A layout (16x4 f32): a.x=(M=ln, K=4*kk+2*hi), a.y=(M=ln, K=4*kk+2*hi+1).
    // Fold scale = 1/sqrt(64) into A once.
    v2f a[16];
#pragma unroll
    for (int kk = 0; kk < 16; ++kk) {
        v2f t = *(const v2f*)(q + (size_t)ln * DK_ + kk * 4 + hi * 2);
        t.x *= 0.125f; t.y *= 0.125f;
        a[kk] = t;
    }

    for (int nt = wave; nt < NT_TILES; nt += 8) {
        const int nb = nt * 16;
        v8f acc = {};
#pragma unroll
        for (int kk = 0; kk < 16; ++kk) {
            // B = K^T: b.x=(K=4*kk+2*hi, N=ln) = Key[nb+ln][4*kk+2*hi]
            v2f bm = *(const v2f*)(k + (size_t)(nb + ln) * DK_ + kk * 4 + hi * 2);
            acc = __builtin_amdgcn_wmma_f32_16x16x4_f32(
                false, a[kk], false, bm, (short)0, acc, false, false);
        }
        // D layout: acc[i] = scores[m = i + 8*hi][n = ln] of this 16x16 tile
#pragma unroll
        for (int i = 0; i < 8; ++i)
            smem[(i + hi * 8) * LDS_STRIDE + nb + ln] = acc[i];
    }
    __syncthreads();

    // ------- Phase 2: mask + row softmax, row resident in VGPRs -------
    // unroll 1: both row iterations share one static code body (I-cache)
#pragma unroll 1
    for (int rr = 0; rr < 2; ++rr) {
        const int  r    = wave + rr * 8;                 // uniform row index
        float*     row  = smem + r * LDS_STRIDE;
        const int* mrow = m + (size_t)r * S_;

        v4f sv[16];
        float mx = -3.4e38f;
#pragma unroll
        for (int it = 0; it < 16; ++it) {
            const int c = lane * 4 + it * 128;
            v4f s  = *(v4f*)(row + c);                   // ds_load_b128
            v4i mk = *(const v4i*)(mrow + c);            // coalesced b128
            s.x = mk.x ? s.x : -1e9f;
            s.y = mk.y ? s.y : -1e9f;
            s.z = mk.z ? s.z : -1e9f;
            s.w = mk.w ? s.w : -1e9f;
            sv[it] = s;
            mx = fmaxf(mx, fmaxf(fmaxf(s.x, s.y), fmaxf(s.z, s.w)));
        }
#pragma unroll
        for (int off = 16; off > 0; off >>= 1) mx = fmaxf(mx, __shfl_xor(mx, off, 32));

        float sum = 0.f;
#pragma unroll
        for (int it = 0; it < 16; ++it) {
            v4f e;
            e.x = __expf(sv[it].x - mx); e.y = __expf(sv[it].y - mx);
            e.z = __expf(sv[it].z - mx); e.w = __expf(sv[it].w - mx);
            sv[it] = e;
            sum += (e.x + e.y) + (e.z + e.w);
        }
#pragma unroll
        for (int off = 16; off > 0; off >>= 1) sum += __shfl_xor(sum, off, 32);
        const float inv = 1.0f / sum;

        float* arow = Attn + ((size_t)bh * S_ + qbase + r) * S_;
#pragma unroll
        for (int it = 0; it < 16; ++it) {
            const int c = lane * 4 + it * 128;
            v4f p = sv[it] * inv;
            *(v4f*)(row + c) = p;                        // P stays in LDS (phase 3)
            __builtin_nontemporal_store(p, (v4f*)(arow + c));  // NT b128 stream
        }
    }
    __syncthreads();

    // ---------- Phase 3: out = P (LDS) x V, split-K over 8 waves ----------
    const int nb   = (wave & 3) * 16;            // N-tile (DK=64 -> 4 tiles)
    const int kbeg = (wave >> 2) * (S_ / 2);     // K half: 0 or 1024
    v8f acc = {};
#pragma unroll 4
    for (int k0 = kbeg; k0 < kbeg + S_ / 2; k0 += 4) {
        // A from LDS: a.x=P[ln][k0+2*hi], a.y=P[ln][k0+2*hi+1]  (ds_load_b64)
        v2f am = *(const v2f*)(smem + (size_t)ln * LDS_STRIDE + k0 + hi * 2);
        // B = V: b.x=V[k0+2*hi][nb+ln], b.y=V[k0+2*hi+1][nb+ln]
        v2f bm;
        bm.x = v[(size_t)(k0 + hi * 2 + 0) * DK_ + nb + ln];
        bm.y = v[(size_t)(k0 + hi * 2 + 1) * DK_ + nb + ln];
        acc = __builtin_amdgcn_wmma_f32_16x16x4_f32(
            false, am, false, bm, (short)0, acc, false, false);
    }

    float* red = smem + QT * LDS_STRIDE;         // 4 x 16 x 17 padded tiles
    if (wave >= 4) {
#pragma unroll
        for (int i = 0; i < 8; ++i)
            red[(wave - 4) * (16 * RED_STRIDE) + (i + hi * 8) * RED_STRIDE + ln] = acc[i];
    }
    __syncthreads();
    if (wave < 4) {
#pragma unroll
        for (int i = 0; i < 8; ++i) {
            const int mr = i + hi * 8;
            float o = acc[i] + red[wave * (16 * RED_STRIDE) + mr * RED_STRIDE + ln];
            Out[((size_t)bh * S_ + qbase + mr) * DK_ + nb + ln] = o;
        }
    }
}

extern "C" void kernel_launch(void* const* d_in, const int* in_sizes, int n_in,
                              void* d_out, int out_size, void* d_ws, size_t ws_size,
                              hipStream_t stream) {
    (void)in_sizes; (void)n_in; (void)d_ws; (void)ws_size; (void)out_size;

    const float* Q = (const float*)d_in[0];
    const float* K = (const float*)d_in[1];
    const float* V = (const float*)d_in[2];
    const int*   M = (const int*)d_in[3];

    float* Out  = (float*)d_out;                                   // [B,H,S,DK]
    float* Attn = (float*)d_out + (size_t)B_ * H_ * S_ * DK_;      // [B,H,S,S]

    const dim3 grid(B_ * H_ * NT_TILES);   // 4096 workgroups
    const dim3 block(256);                 // 8 waves (wave32)
    const size_t smem_bytes =
        ((size_t)QT * LDS_STRIDE + RED_FLOATS) * sizeof(float);    // ~135.7 KB

    ScaledDotProductAttention_42090679501324_kernel<<<grid, block, smem_bytes, stream>>>(
        Q, K, V, M, Out, Attn);
}